// AttentionStyleEstimator_12378095747883
// MI455X (gfx1250) — compile-verified
//
#include <hip/hip_runtime.h>
#include <hip/hip_bf16.h>

// ---------------------------------------------------------------------------
// AttentionStyleEstimator for MI455X (gfx1250, wave32).
//   Stage 1: Q = (x@Wq)*SCALE, K = x@Wk   via V_WMMA_F32_16X16X4_F32
//   Stage 2: scores = Q·K^T (WMMA), exact top-64 threshold per row via
//            4-pass radix select on order-preserving uint bits, softmax.
// ---------------------------------------------------------------------------

typedef __attribute__((ext_vector_type(2))) float v2f;
typedef __attribute__((ext_vector_type(8))) float v8f;

#define WMMA4(a, b, c) \
  __builtin_amdgcn_wmma_f32_16x16x4_f32(false, (a), false, (b), (short)0, (c), false, false)

__device__ __forceinline__ unsigned f2ord(float f) {
  unsigned u = __float_as_uint(f);
  return (u & 0x80000000u) ? ~u : (u | 0x80000000u);
}
__device__ __forceinline__ float ord2f(unsigned u) {
  unsigned b = (u & 0x80000000u) ? (u & 0x7fffffffu) : ~u;
  return __uint_as_float(b);
}
__device__ __forceinline__ unsigned umaxu(unsigned a, unsigned b) { return a > b ? a : b; }

// ---------------------------------------------------------------------------
// Stage 1: projections. Grid 512 x 256 threads. Each wave: one 16x64 output
// stripe of one of the two GEMMs (4096 wave-tasks total).
// Output layout: [b][h][n][64] so stage-2 rows are contiguous 64-float runs.
// ---------------------------------------------------------------------------
__global__ __launch_bounds__(256) void proj_kernel(
    const float* __restrict__ x, const float* __restrict__ Wq,
    const float* __restrict__ Wk, float* __restrict__ Qb, float* __restrict__ Kb) {
  const int lane = threadIdx.x & 31;
  const int wv   = threadIdx.x >> 5;
  const int g    = blockIdx.x * 8 + wv;   // 0..4095
  const int mat  = g >> 11;               // 0 = Q, 1 = K
  const int rem  = g & 2047;
  const int m    = rem >> 3;              // 16-row M tile, 0..255
  const int n0   = (rem & 7) * 64;        // 64-wide N stripe

  const float* W = mat ? Wk : Wq;
  float*       O = mat ? Kb : Qb;
  const float  scale = mat ? 1.0f : 0.125f;   // SCALE = DIM_HEAD^-0.5 folded into Q

  const int mrow  = lane & 15;
  const int khalf = (lane >> 4) << 1;     // 0 or 2

  v8f acc[4];
#pragma unroll
  for (int t = 0; t < 4; ++t) acc[t] = (v8f){0.f, 0.f, 0.f, 0.f, 0.f, 0.f, 0.f, 0.f};

  const float* xrow = x + (size_t)(m * 16 + mrow) * 1024;
  for (int k0 = 0; k0 < 1024; k0 += 4) {
    v2f a = *(const v2f*)(xrow + k0 + khalf);                   // A[M=mrow][k0+khalf..+1]
    const float* wp = W + (size_t)(k0 + khalf) * 512 + n0 + mrow;
#pragma unroll
    for (int t = 0; t < 4; ++t) {
      v2f b;                                                    // B[k][n] = W[k][n]
      b[0] = wp[t * 16];
      b[1] = wp[512 + t * 16];
      acc[t] = WMMA4(a, b, acc[t]);
    }
  }

  const int Mb = (lane < 16) ? 0 : 8;
  const int Nn = lane & 15;
  const int h  = n0 >> 6;
#pragma unroll
  for (int t = 0; t < 4; ++t) {
    const int d = t * 16 + Nn;
#pragma unroll
    for (int i = 0; i < 8; ++i) {
      const int R = m * 16 + Mb + i;
      const int b = R >> 11, n = R & 2047;
      O[((size_t)(b * 8 + h) * 2048 + n) * 64 + d] = acc[t][i] * scale;
    }
  }
}

// ---------------------------------------------------------------------------
// Stage 2: scores + top-64 mask + softmax. Grid 2048 x 256 threads.
// Block = (bh, 16 q-rows). 8 waves compute 16x16 tiles into a 16x256 LDS
// chunk buffer; each wave owns 2 rows (64 ordered-uint scores per lane).
// ---------------------------------------------------------------------------
#define ROWPAD 260   // 260 mod 64 = 4  ->  rows i and i+8 hit disjoint banks

__global__ __launch_bounds__(256) void attn_kernel(
    const float* __restrict__ Qb, const float* __restrict__ Kb,
    float* __restrict__ out) {
  __shared__ float    cb[16 * ROWPAD];    // 16.6 KB score staging
  __shared__ unsigned hist[8 * 256];      // per-wave radix histograms (8 KB)

  const int tid = threadIdx.x, lane = tid & 31, wv = tid >> 5;
  const int bh = blockIdx.x >> 7;
  const int q0 = (blockIdx.x & 127) << 4;

  const int mrow  = lane & 15;
  const int khalf = (lane >> 4) << 1;

  // A fragments for the whole 16x64 Q tile (reused by all 128 key tiles)
  const float* qrow = Qb + ((size_t)bh * 2048 + q0 + mrow) * 64;
  v2f a[16];
#pragma unroll
  for (int kk = 0; kk < 16; ++kk) a[kk] = *(const v2f*)(qrow + kk * 4 + khalf);

  unsigned u0[64], u1[64];    // this wave's two rows, ordered-uint score bits
  const float* kbase = Kb + (size_t)bh * 2048 * 64;

#pragma unroll
  for (int c = 0; c < 8; ++c) {           // key chunks of 256
#pragma unroll
    for (int tt = 0; tt < 2; ++tt) {      // 2 of the chunk's 16 tiles per wave
      const int keyl = (wv * 2 + tt) * 16;
      const float* krow = kbase + (size_t)(c * 256 + keyl + mrow) * 64;
      v8f acc = (v8f){0.f, 0.f, 0.f, 0.f, 0.f, 0.f, 0.f, 0.f};
#pragma unroll
      for (int kk = 0; kk < 16; ++kk) {   // B[k][n] = K[key+n][k] -> float2 of k
        v2f b = *(const v2f*)(krow + kk * 4 + khalf);
        acc = WMMA4(a[kk], b, acc);
      }
      const int Mb = (lane < 16) ? 0 : 8;
      const int Nn = lane & 15;
#pragma unroll
      for (int i = 0; i < 8; ++i) cb[(Mb + i) * ROWPAD + keyl + Nn] = acc[i];
    }
    __syncthreads();
    // pull my 2 rows' slice of this chunk into registers
#pragma unroll
    for (int s = 0; s < 2; ++s) {
      const float4 v0 = *(const float4*)(cb + (wv * 2 + 0) * ROWPAD + 4 * lane + 128 * s);
      const float4 v1 = *(const float4*)(cb + (wv * 2 + 1) * ROWPAD + 4 * lane + 128 * s);
      const int jb = c * 8 + s * 4;
      u0[jb + 0] = f2ord(v0.x); u0[jb + 1] = f2ord(v0.y);
      u0[jb + 2] = f2ord(v0.z); u0[jb + 3] = f2ord(v0.w);
      u1[jb + 0] = f2ord(v1.x); u1[jb + 1] = f2ord(v1.y);
      u1[jb + 2] = f2ord(v1.z); u1[jb + 3] = f2ord(v1.w);
    }
    __syncthreads();                      // cb reused next chunk
  }

  unsigned mu0 = 0, mu1 = 0;
#pragma unroll
  for (int j = 0; j < 64; ++j) { mu0 = umaxu(mu0, u0[j]); mu1 = umaxu(mu1, u1[j]); }

  auto process_row = [&](unsigned (&u)[64], unsigned mu, int r) {
    // wave max (ordered-uint max == float max)
#pragma unroll
    for (int off = 16; off > 0; off >>= 1)
      mu = umaxu(mu, (unsigned)__shfl_xor((int)mu, off, 32));
    const float mx = ord2f(mu);

    // exact radix select of 64th-largest over 2048 values (ties kept)
    unsigned prefix = 0, need = 64;
    unsigned* h = &hist[wv * 256];
#pragma unroll
    for (int pass = 0; pass < 4; ++pass) {
      const int shift = 24 - 8 * pass;
#pragma unroll
      for (int j = 0; j < 8; ++j) h[lane * 8 + j] = 0u;
      __syncthreads();
#pragma unroll
      for (int j = 0; j < 64; ++j) {
        const unsigned uu = u[j];
        const bool match = (pass == 0) || ((uu >> (shift + 8)) == (prefix >> (shift + 8)));
        if (match) atomicAdd(&h[(uu >> shift) & 255u], 1u);
      }
      __syncthreads();
      unsigned cnt[8], ls = 0;
#pragma unroll
      for (int j = 0; j < 8; ++j) { cnt[j] = h[lane * 8 + j]; ls += cnt[j]; }
      unsigned higher = 0;                        // sum of bins owned by higher lanes
#pragma unroll
      for (int j = 0; j < 32; ++j) {
        const unsigned v = (unsigned)__shfl((int)ls, j, 32);
        if (j > lane) higher += v;
      }
      unsigned cum = higher, sneed = 0;
      int selb = -1;
#pragma unroll
      for (int j = 7; j >= 0; --j) {              // scan my 8 bins from the top
        const unsigned nc = cum + cnt[j];
        if (selb < 0 && cum < need && nc >= need) { selb = lane * 8 + j; sneed = need - cum; }
        cum = nc;
      }
      const unsigned long long bal = __ballot(selb >= 0);
      const int winner = __ffsll(bal) - 1;
      selb = __shfl(selb, winner, 32);
      need = (unsigned)__shfl((int)sneed, winner, 32);
      prefix |= ((unsigned)selb) << shift;
      __syncthreads();
    }
    const unsigned T = prefix;                    // bits of min_k; keep u >= T

    // masked softmax (exp stored in place of u)
    float s = 0.0f;
#pragma unroll
    for (int j = 0; j < 64; ++j) {
      const float f = ord2f(u[j]);
      const float e = (u[j] >= T) ? __expf(f - mx) : 0.0f;
      u[j] = __float_as_uint(e);
      s += e;
    }
#pragma unroll
    for (int off = 16; off > 0; off >>= 1) s += __shfl_xor(s, off, 32);
    const float inv = 1.0f / s;

    float* orow = out + ((size_t)bh * 2048 + (size_t)(q0 + r)) * 2048;
#pragma unroll
    for (int c = 0; c < 8; ++c) {
#pragma unroll
      for (int s2 = 0; s2 < 2; ++s2) {
        const int jb = c * 8 + s2 * 4;
        float4 v;
        v.x = __uint_as_float(u[jb + 0]) * inv;
        v.y = __uint_as_float(u[jb + 1]) * inv;
        v.z = __uint_as_float(u[jb + 2]) * inv;
        v.w = __uint_as_float(u[jb + 3]) * inv;
        *(float4*)(orow + c * 256 + 4 * lane + 128 * s2) = v;   // coalesced b128
      }
    }
  };

  process_row(u0, mu0, 2 * wv + 0);
  process_row(u1, mu1, 2 * wv + 1);
}

// ---------------------------------------------------------------------------
extern "C" void kernel_launch(void* const* d_in, const int* in_sizes, int n_in,
                              void* d_out, int out_size, void* d_ws, size_t ws_size,
                              hipStream_t stream) {
  const float* x  = (const float*)d_in[0];
  const float* Wq = (const float*)d_in[1];
  const float* Wk = (const float*)d_in[2];
  float* out = (float*)d_out;

  float* Qb = (float*)d_ws;                       // [2*8][2048][64] = 8 MB
  float* Kb = Qb + (size_t)2 * 8 * 2048 * 64;     // + 8 MB

  proj_kernel<<<dim3(512), dim3(256), 0, stream>>>(x, Wq, Wk, Qb, Kb);
  attn_kernel<<<dim3(2048), dim3(256), 0, stream>>>(Qb, Kb, out);
}